// WeightedAggEdgeConcatNode_67439576482328
// MI455X (gfx1250) — compile-verified
//
#include <hip/hip_runtime.h>
#include <hip/hip_bf16.h>
#include <stdint.h>

// Problem constants (match reference)
#define N_NODES  50000
#define N_EDGES  800000
#define NODE_DIM 64
#define EDGE_DIM 64
#define IN_DIM   128   // NODE_DIM + EDGE_DIM
#define OUT_DIM  128

typedef __attribute__((ext_vector_type(2))) float v2f;
typedef __attribute__((ext_vector_type(8))) float v8f;

// ---------------------------------------------------------------------------
// Kernel 1: zero the [N_NODES, IN_DIM] accumulator in workspace (float4 wide)
// ---------------------------------------------------------------------------
__global__ void zero_accum_kernel(float4* __restrict__ p, int n4) {
    int i = blockIdx.x * blockDim.x + threadIdx.x;
    if (i < n4) p[i] = make_float4(0.f, 0.f, 0.f, 0.f);
}

// ---------------------------------------------------------------------------
// Kernel 2: edge scatter.  One wave32 per edge.
//   lanes 0-15 : move h[src][0..63]        -> accum[dst][0..63]
//   lanes 16-31: move edge_labels[e][0..63]-> accum[dst][64..127]
// Each lane does one float4 (b128) load + 4 global_atomic_add_f32.
// h (12.8MB) and accum (25.6MB) stay L2-resident; edge_labels streams.
// ---------------------------------------------------------------------------
__global__ void scatter_kernel(const float* __restrict__ h,
                               const float* __restrict__ el,
                               const long long* __restrict__ src,
                               const long long* __restrict__ dst,
                               float* __restrict__ accum) {
    int edge = (blockIdx.x * blockDim.x + threadIdx.x) >> 5;
    int lane = threadIdx.x & 31;
    if (edge >= N_EDGES) return;

    long long s = src[edge];
    long long d = dst[edge];

    int half = lane >> 4;       // 0 = node half, 1 = edge half
    int lm   = lane & 15;       // 16 lanes * 4 floats = 64 floats per half

    const float* sp = half ? (el + (size_t)edge * EDGE_DIM)
                           : (h  + (size_t)s    * NODE_DIM);
    const float4 v = *(const float4*)(sp + lm * 4);

    float* dp = accum + (size_t)d * IN_DIM + half * NODE_DIM + lm * 4;
    atomicAdd(dp + 0, v.x);
    atomicAdd(dp + 1, v.y);
    atomicAdd(dp + 2, v.z);
    atomicAdd(dp + 3, v.w);
}

// ---------------------------------------------------------------------------
// Kernel 3: z = elu(accum @ W) via V_WMMA_F32_16X16X4_F32.
// One wave per 16x16 output tile; K = 128 -> 32 unrolled WMMA steps.
//
// f32 fragment layouts (ISA 7.12.2):
//   A 16x4 : lanes 0-15 M=0..15 {K=k, k+1}; lanes 16-31 M=0..15 {K=k+2, k+3}
//   B 4x16 : lanes 0-15 N=0..15 {K=k, k+1}; lanes 16-31 N=0..15 {K=k+2, k+3}
//   C/D    : VGPR j = row (j + 8*half), col = lane%16
// ---------------------------------------------------------------------------
__global__ void gemm_elu_kernel(const float* __restrict__ A,   // [N_NODES, 128]
                                const float* __restrict__ W,   // [128, 128] row-major [k][n]
                                float* __restrict__ out) {     // [N_NODES, 128]
    const int wavesPerBlock = blockDim.x >> 5;
    int wave = blockIdx.x * wavesPerBlock + (threadIdx.x >> 5);
    int lane = threadIdx.x & 31;

    const int nTiles = OUT_DIM / 16;            // 8
    int mTile = wave / nTiles;
    int nTile = wave % nTiles;
    if (mTile * 16 >= N_NODES) return;          // wave-uniform: EXEC all-1s for WMMA

    int half = lane >> 4;
    int lm   = lane & 15;

    // A fragment base: row (mTile*16+lm), K offset 2*half (float2-aligned)
    const float* aPtr = A + (size_t)(mTile * 16 + lm) * IN_DIM + 2 * half;
    // B fragment base: col (nTile*16+lm), K offset 2*half
    const float* bPtr = W + (size_t)(2 * half) * OUT_DIM + nTile * 16 + lm;

    v8f c = {0.f, 0.f, 0.f, 0.f, 0.f, 0.f, 0.f, 0.f};

#pragma unroll
    for (int k = 0; k < IN_DIM; k += 4) {
        float2 av = *(const float2*)(aPtr + k);          // K-contiguous: single b64 load
        v2f a; a.x = av.x; a.y = av.y;
        v2f b;
        b.x = bPtr[(size_t)(k + 0) * OUT_DIM];
        b.y = bPtr[(size_t)(k + 1) * OUT_DIM];
        // 8 args: (neg_a, A, neg_b, B, c_mod, C, reuse_a, reuse_b)
        c = __builtin_amdgcn_wmma_f32_16x16x4_f32(false, a, false, b,
                                                  (short)0, c, false, false);
    }

    // Epilogue: ELU + scattered per-row stores (row stride OUT_DIM)
    float* oBase = out + (size_t)(mTile * 16) * OUT_DIM + nTile * 16 + lm;
#pragma unroll
    for (int j = 0; j < 8; ++j) {
        int row = j + half * 8;
        float x = c[j];
        float e = x > 0.f ? x : (__expf(x) - 1.0f);
        oBase[(size_t)row * OUT_DIM] = e;
    }
}

// ---------------------------------------------------------------------------
extern "C" void kernel_launch(void* const* d_in, const int* in_sizes, int n_in,
                              void* d_out, int out_size, void* d_ws, size_t ws_size,
                              hipStream_t stream) {
    const float*     h   = (const float*)d_in[0];       // [N_NODES, 64]
    const float*     el  = (const float*)d_in[1];       // [N_EDGES, 64]
    const float*     W   = (const float*)d_in[2];       // [128, 128]
    const long long* src = (const long long*)d_in[3];   // [N_EDGES] int64
    const long long* dst = (const long long*)d_in[4];   // [N_EDGES] int64
    float* out   = (float*)d_out;                       // [N_NODES, 128]
    float* accum = (float*)d_ws;                        // [N_NODES, 128] scratch

    // 1) zero accumulator (must happen every call: graph replay re-adds)
    int n4 = N_NODES * IN_DIM / 4;                      // 1.6M float4
    zero_accum_kernel<<<(n4 + 255) / 256, 256, 0, stream>>>((float4*)accum, n4);

    // 2) scatter-add: one wave per edge -> 800000 waves
    {
        long long threads = (long long)N_EDGES * 32;
        int blocks = (int)((threads + 255) / 256);      // 100000
        scatter_kernel<<<blocks, 256, 0, stream>>>(h, el, src, dst, accum);
    }

    // 3) WMMA GEMM + ELU: 3125 * 8 = 25000 tiles, 8 waves/block -> 3125 blocks
    {
        int tiles  = (N_NODES / 16) * (OUT_DIM / 16);
        int blocks = tiles / 8;
        gemm_elu_kernel<<<blocks, 256, 0, stream>>>(accum, W, out);
    }
}